// RealEmbedding_39024072851809
// MI455X (gfx1250) — compile-verified
//
#include <hip/hip_runtime.h>

// Problem constants (from reference): B=64, T=512, C_IN=128, C_OUT=16
#define ROWS       32768      // B*T
#define CIN        128
#define COUT       16
#define OUTW       2048       // CIN*COUT
#define WAVES_PER_WG 4
#define ROWS_PER_WG  (WAVES_PER_WG * 16)   // 64
#define XS_ROW_STRIDE 132                  // 128 + 4 pad dwords: conflict-free column reads, 16B-aligned rows
#define XS_WAVE_BLOCK (16 * XS_ROW_STRIDE) // 2112 floats per wave tile

typedef float v2f __attribute__((ext_vector_type(2)));
typedef float v8f __attribute__((ext_vector_type(8)));

__global__ __launch_bounds__(128) void
mulbias_wmma_kernel(const float* __restrict__ x,
                    const float* __restrict__ W,
                    const float* __restrict__ bias,
                    float* __restrict__ out) {
    // wb: interleaved [i][0..15] = W row i, [i][16..31] = bias row i (64B apart -> ds_load_2addr)
    __shared__ float wb[CIN * 32];                         // 16 KiB
    __shared__ float xs[WAVES_PER_WG * XS_WAVE_BLOCK];     // 33 KiB (padded x tile)

    const int tid  = threadIdx.x;
    const int wave = tid >> 5;
    const int lane = tid & 31;
    const int rowbase = blockIdx.x * ROWS_PER_WG;

    // ---- Stage W and bias interleaved into LDS (coalesced b128) ----
    {
        const float4* W4 = (const float4*)W;
        const float4* B4 = (const float4*)bias;
        float4* wb4 = (float4*)wb;
        #pragma unroll
        for (int k = tid; k < (CIN * COUT) / 4; k += 128) {
            const int i  = k >> 2;      // feature row
            const int c4 = k & 3;       // float4 within row
            wb4[i * 8 + c4]     = W4[k];
            wb4[i * 8 + 4 + c4] = B4[k];
        }
    }

    // ---- Stage x tile (64 rows x 128 cols) into padded LDS, coalesced b128 ----
    {
        const float4* x4 = (const float4*)(x + (size_t)rowbase * CIN);
        #pragma unroll
        for (int k = tid; k < (ROWS_PER_WG * CIN) / 4; k += 128) {
            const int r  = k >> 5;      // 32 float4 per row
            const int c4 = k & 31;
            float4 v = x4[k];
            const int idx = (r >> 4) * XS_WAVE_BLOCK + (r & 15) * XS_ROW_STRIDE + (c4 << 2);
            *(float4*)(xs + idx) = v;   // row stride 528B, block stride 8448B -> 16B aligned
        }
    }
    __syncthreads();

    // ---- Per-wave: 16-row tile, iterate over all 128 input features ----
    const float* xw   = xs + wave * XS_WAVE_BLOCK;
    float* outw = out + (size_t)(rowbase + wave * 16) * OUTW;

    const int  l16 = lane & 15;
    const bool lo  = (lane < 16);
    const int  rowoff = lo ? 0 : 8;    // D layout: lanes 0-15 hold M=v, lanes 16-31 hold M=v+8

    for (int i = 0; i < CIN; ++i) {
        // A (16x4 f32): K=0 column = x[row, i] (VGPR0 lanes 0-15), K=1 column = 1.0 (VGPR1).
        // Lanes 16-31 hold K=2,3; those are nullified by B rows 2,3 being zero
        // (duplicated x values are finite -> no 0*Inf NaN hazard).
        v2f a;
        a.x = xw[l16 * XS_ROW_STRIDE + i];   // banks (4*lane + i) % 64 -> conflict-free
        a.y = 1.0f;                          // loop-invariant, no mask needed

        // B (4x16 f32): row K=0 = W[i,:] (VGPR0 lanes 0-15), row K=1 = bias[i,:] (VGPR1
        // lanes 0-15); rows K=2,3 (lanes 16-31) forced to zero.
        float wv = wb[i * 32 + l16];         // fuses into one ds_load_2addr_b32
        float bv = wb[i * 32 + 16 + l16];
        v2f bm;
        bm.x = lo ? wv : 0.0f;
        bm.y = lo ? bv : 0.0f;

        // D = x_col * W_row + 1 * bias_row, C = inline 0
        v8f c = {0.f, 0.f, 0.f, 0.f, 0.f, 0.f, 0.f, 0.f};
        v8f d = __builtin_amdgcn_wmma_f32_16x16x4_f32(
            /*neg_a=*/false, a, /*neg_b=*/false, bm,
            /*c_mod=*/(short)0, c, /*reuse_a=*/false, /*reuse_b=*/false);

        // Store 16x16 tile: each b32 store = two contiguous 64B segments (rows v and v+8)
        float* op = outw + i * COUT + l16;
        #pragma unroll
        for (int v = 0; v < 8; ++v) {
            op[(size_t)(rowoff + v) * OUTW] = d[v];
        }
    }
}

extern "C" void kernel_launch(void* const* d_in, const int* in_sizes, int n_in,
                              void* d_out, int out_size, void* d_ws, size_t ws_size,
                              hipStream_t stream) {
    const float* x    = (const float*)d_in[0];
    const float* W    = (const float*)d_in[1];
    const float* bias = (const float*)d_in[2];
    float* out = (float*)d_out;

    dim3 grid(ROWS / ROWS_PER_WG);  // 512 workgroups
    dim3 block(128);                // 4 waves (wave32)
    mulbias_wmma_kernel<<<grid, block, 0, stream>>>(x, W, bias, out);
}